// _FreqAttentionModule_2894807957851
// MI455X (gfx1250) — compile-verified
//
#include <hip/hip_runtime.h>
#include <hip/hip_bf16.h>
#include <stdint.h>

// CDNA5 / gfx1250 single-pass kernel:
//   out[b,c,n] = alpha * (1/512) * sum_d x[b,d,n] + x[b,c,n]
// (the reference's reversed-softmax attention collapses to uniform 1/C)

typedef __attribute__((ext_vector_type(2))) float v2f;
typedef __attribute__((ext_vector_type(4))) float v4f;
typedef __attribute__((ext_vector_type(8))) float v8f;

#define BB 32
#define CC 512
#define NN 4096
#define NT 128           // n-tile per workgroup
#define ROWP 132         // padded LDS row length in floats (132*4 = 528 B, 16B-aligned)
#define NTHREADS 512

// 16-byte global -> LDS asynchronous copy (ASYNCcnt-tracked, bypasses VGPRs)
__device__ __forceinline__ void async_cp16(uint32_t lds_byte_addr, uint64_t global_addr) {
    asm volatile("global_load_async_to_lds_b128 %0, %1, off"
                 :: "v"(lds_byte_addr), "v"(global_addr)
                 : "memory");
}

__device__ __forceinline__ void wait_async0() {
    asm volatile("s_wait_asynccnt 0" ::: "memory");
}

__global__ __launch_bounds__(NTHREADS, 1)
void freq_attn_kernel(const float* __restrict__ x,
                      const float* __restrict__ alpha,
                      float* __restrict__ out) {
    extern __shared__ float smem[];          // [CC][ROWP] tile + [NT] sums
    float* sums = smem + CC * ROWP;

    const int t    = threadIdx.x;
    const int blk  = blockIdx.x;
    const int b    = blk >> 5;               // 32 n-tiles per batch
    const int tile = blk & 31;
    const int n0   = tile * NT;

    const float* xb = x + ((size_t)b * CC) * NN + n0;   // row start for c=0

    // ---------------- Phase 1: async DMA tile into LDS ----------------
    // 512 rows x 8 chunks of 16B = 4096 chunks; 8 per thread.
    const uint32_t lds_base = (uint32_t)(uintptr_t)smem;
    #pragma unroll
    for (int i = 0; i < 8; ++i) {
        int idx = i * NTHREADS + t;          // 0..4095
        int c   = idx >> 3;
        int k   = idx & 7;                   // 16B chunk within row
        uint64_t g = (uint64_t)(uintptr_t)(xb + (size_t)c * NN + k * 4);
        uint32_t l = lds_base + (uint32_t)((c * ROWP + k * 4) * 4);
        async_cp16(l, g);
    }
    wait_async0();
    __syncthreads();

    // ---------------- Phase 2: channel sums via WMMA ----------------
    // D = A(ones,16x4) x B(x-chunk,4x16) accumulated over c -> every row of D
    // holds sum_c x[b,c,n] for the wave's 16 columns.
    const int w    = t >> 5;
    const int lane = t & 31;
    if (w < (NT / 16)) {                     // 8 waves do WMMA, uniform per wave
        const int nbase  = w * 16;
        const int rowoff = lane >> 4;        // 0 or 1 (K slot half)
        const int ncol   = nbase + (lane & 15);
        v2f a; a[0] = 1.0f; a[1] = 1.0f;     // ones A-matrix
        v8f acc = {};
        #pragma unroll 4
        for (int c = 0; c < CC; c += 4) {
            v2f bm;
            bm[0] = smem[(c + rowoff)     * ROWP + ncol];
            bm[1] = smem[(c + 2 + rowoff) * ROWP + ncol];
            acc = __builtin_amdgcn_wmma_f32_16x16x4_f32(
                      false, a, false, bm, (short)0, acc, false, false);
        }
        if (lane < 16) sums[nbase + lane] = acc[0];   // row M=0 of D
    }
    __syncthreads();

    // ---------------- Phase 3: out = (alpha/512)*sum + x ----------------
    const float ascale = alpha[0] * (1.0f / (float)CC);
    float* outb = out + ((size_t)b * CC) * NN + n0;
    const v4f* sums4 = (const v4f*)sums;
    #pragma unroll 4
    for (int i = 0; i < 32; ++i) {
        int idx = i * NTHREADS + t;          // 0..16383 vec4 slots
        int c   = idx >> 5;                  // 32 vec4 per row
        int v   = idx & 31;
        v4f xv = *(const v4f*)(smem + c * ROWP + v * 4);
        v4f sv = sums4[v];
        v4f o;
        o.x = fmaf(ascale, sv.x, xv.x);
        o.y = fmaf(ascale, sv.y, xv.y);
        o.z = fmaf(ascale, sv.z, xv.z);
        o.w = fmaf(ascale, sv.w, xv.w);
        __builtin_nontemporal_store(o, (v4f*)(outb + (size_t)c * NN + v * 4));
    }
}

extern "C" void kernel_launch(void* const* d_in, const int* in_sizes, int n_in,
                              void* d_out, int out_size, void* d_ws, size_t ws_size,
                              hipStream_t stream) {
    (void)in_sizes; (void)n_in; (void)d_ws; (void)ws_size; (void)out_size;
    const float* x     = (const float*)d_in[0];
    // d_in[1] (fre) provably does not affect the output — attention collapses
    // to uniform 1/C because feat_f rows are identical across channels.
    const float* alpha = (const float*)d_in[2];
    float* out = (float*)d_out;

    const size_t shmem = (size_t)(CC * ROWP + NT) * sizeof(float);  // ~264.5 KB < 320 KB/WGP
    (void)hipFuncSetAttribute((const void*)freq_attn_kernel,
                              hipFuncAttributeMaxDynamicSharedMemorySize,
                              (int)shmem);

    dim3 grid(BB * (NN / NT));               // 32 batches x 32 n-tiles = 1024 WGs
    dim3 block(NTHREADS);                    // 16 wave32s
    freq_attn_kernel<<<grid, block, shmem, stream>>>(x, alpha, out);
}